// GCN_21242908246158
// MI455X (gfx1250) — compile-verified
//
#include <hip/hip_runtime.h>

typedef float v2f __attribute__((ext_vector_type(2)));
typedef float v8f __attribute__((ext_vector_type(8)));

#define DFEAT 256
#define TILE_K 64

__device__ __forceinline__ void atomic_add_f32(float* p, float v) {
    // non-returning fadd, agent scope -> global_atomic_add_f32 (no CAS loop)
    __hip_atomic_fetch_add(p, v, __ATOMIC_RELAXED, __HIP_MEMORY_SCOPE_AGENT);
}

// ---------------------------------------------------------------- degrees
__global__ void gcn_init_deg(float* __restrict__ deg_out,
                             float* __restrict__ deg_in, int n) {
    int i = blockIdx.x * blockDim.x + threadIdx.x;
    if (i < n) { deg_out[i] = 1.0f; deg_in[i] = 1.0f; }   // self-loop
}

__global__ void gcn_count_deg(const int* __restrict__ src,
                              const int* __restrict__ dst,
                              float* __restrict__ deg_out,
                              float* __restrict__ deg_in, int e) {
    int i = blockIdx.x * blockDim.x + threadIdx.x;
    if (i < e) {
        atomic_add_f32(&deg_out[src[i]], 1.0f);
        atomic_add_f32(&deg_in[dst[i]], 1.0f);
    }
}

__global__ void gcn_make_norm(float* __restrict__ ns,
                              float* __restrict__ nd, int n) {
    int i = blockIdx.x * blockDim.x + threadIdx.x;
    if (i < n) {
        ns[i] = 1.0f / sqrtf(ns[i]);   // deg >= 1 always (self loop)
        nd[i] = 1.0f / sqrtf(nd[i]);
    }
}

// ---------------------------------------------------------------- WMMA GEMM
// h = (X * ns[row]) @ W ; also writes agg = h (self-loop init for scatter).
// Block: 128 threads = 4 waves; tile: 16 rows x 64 cols; K staged at 64.
// fp32 WMMA layout (ISA 7.12.2): lane l -> m=n=l&15, hk=l>>4;
//   A frag = As[m][k+2*hk .. +1], B frag = Ws[k+2*hk][n], Ws[k+2*hk+1][n];
//   C/D VGPR r -> row r + 8*hk.
__global__ __launch_bounds__(128)
void gcn_gemm_wmma(const float* __restrict__ X,    // [N,256]
                   const float* __restrict__ ns,   // [N]
                   const float* __restrict__ W,    // [256,256]
                   float* __restrict__ h,          // [N,256]
                   float* __restrict__ agg,        // [N,256] (init = h)
                   int n) {
    __shared__ float As[16][TILE_K + 4];   // stride 68 -> bank stride 4, 16B-aligned rows
    __shared__ float Ws[TILE_K][64];

    const int row0 = blockIdx.x * 16;
    const int col0 = blockIdx.y * 64;
    const int tid  = threadIdx.x;
    const int wave = tid >> 5;        // 0..3 : 16-col subtile
    const int lane = tid & 31;
    const int m    = lane & 15;       // A row == B/C col within tile
    const int hk   = lane >> 4;       // K-half select (ISA fp32 layout)

    v8f acc = {0.f, 0.f, 0.f, 0.f, 0.f, 0.f, 0.f, 0.f};

    for (int kc = 0; kc < DFEAT; kc += TILE_K) {
        // ---- stage A tile (16 x 64) as float4, scaled by norm_src
        {
            const int rr = tid >> 3;          // 0..15 : row
            const int cb = tid & 7;           // float4 column base
            const int rg = row0 + rr;
            const int rc = (rg < n) ? rg : 0;
            const float s = (rg < n) ? ns[rc] : 0.0f;
            const float4* xrow = (const float4*)(X + (size_t)rc * DFEAT + kc);
#pragma unroll
            for (int i = 0; i < 2; ++i) {
                const int c4 = cb + i * 8;    // 0..15
                float4 v = xrow[c4];
                v.x *= s; v.y *= s; v.z *= s; v.w *= s;
                *(float4*)&As[rr][c4 * 4] = v;
            }
        }
        // ---- stage W tile (64 x 64) as float4
        {
            const int c4 = tid & 15;          // float4 column
            const int rb = tid >> 4;          // 0..7
#pragma unroll
            for (int i = 0; i < 8; ++i) {
                const int rr = rb + i * 8;    // 0..63
                *(float4*)&Ws[rr][c4 * 4] =
                    *(const float4*)(W + (size_t)(kc + rr) * DFEAT + col0 + c4 * 4);
            }
        }
        __syncthreads();

#pragma unroll
        for (int k4 = 0; k4 < TILE_K; k4 += 4) {
            v2f a = *(const v2f*)&As[m][k4 + 2 * hk];
            v2f b;
            b.x = Ws[k4 + 2 * hk + 0][wave * 16 + m];
            b.y = Ws[k4 + 2 * hk + 1][wave * 16 + m];
            acc = __builtin_amdgcn_wmma_f32_16x16x4_f32(
                false, a, false, b, (short)0, acc, false, false);
        }
        __syncthreads();
    }

    // ---- epilogue: dual store (h and agg); C/D layout: VGPR r -> row r+8*hk
    const int col = col0 + wave * 16 + m;
#pragma unroll
    for (int r = 0; r < 8; ++r) {
        const int row = row0 + r + 8 * hk;
        if (row < n) {
            const float v = acc[r];
            h[(size_t)row * DFEAT + col]   = v;
            agg[(size_t)row * DFEAT + col] = v;
        }
    }
}

// ---------------------------------------------------------------- SpMM scatter
// 4 edges per 256-thread block; 64 lanes/edge, float4 per lane.
__global__ __launch_bounds__(256)
void gcn_scatter(const float* __restrict__ h, float* __restrict__ agg,
                 const int* __restrict__ src, const int* __restrict__ dst,
                 int e) {
    const int edge = blockIdx.x * 4 + (threadIdx.x >> 6);
    if (edge >= e) return;
    const int t = threadIdx.x & 63;
    const int s = src[edge];
    const int d = dst[edge];
    const float4 v = ((const float4*)(h + (size_t)s * DFEAT))[t];
    float* base = agg + (size_t)d * DFEAT + t * 4;
    atomic_add_f32(base + 0, v.x);
    atomic_add_f32(base + 1, v.y);
    atomic_add_f32(base + 2, v.z);
    atomic_add_f32(base + 3, v.w);
}

// ---------------------------------------------------------------- epilogue
// x = [relu](x * nd[row] + b[col]) in place, float4 wide.
__global__ void gcn_bias_act(float* __restrict__ x, const float* __restrict__ nd,
                             const float* __restrict__ bias, int relu,
                             long total4) {
    const long i = (long)blockIdx.x * blockDim.x + threadIdx.x;
    if (i >= total4) return;
    const int row = (int)(i >> 6);        // 64 float4 per row
    const int c4  = (int)(i & 63) * 4;
    const float s = nd[row];
    float4 v = ((float4*)x)[i];
    const float4 b = *(const float4*)(bias + c4);
    v.x = v.x * s + b.x;
    v.y = v.y * s + b.y;
    v.z = v.z * s + b.z;
    v.w = v.w * s + b.w;
    if (relu) {
        v.x = fmaxf(v.x, 0.f); v.y = fmaxf(v.y, 0.f);
        v.z = fmaxf(v.z, 0.f); v.w = fmaxf(v.w, 0.f);
    }
    ((float4*)x)[i] = v;
}

// ---------------------------------------------------------------- launch
extern "C" void kernel_launch(void* const* d_in, const int* in_sizes, int n_in,
                              void* d_out, int out_size, void* d_ws, size_t ws_size,
                              hipStream_t stream) {
    const float* features = (const float*)d_in[0];
    const int*   src      = (const int*)d_in[1];
    const int*   dst      = (const int*)d_in[2];
    const float* W1       = (const float*)d_in[3];
    const float* b1       = (const float*)d_in[4];
    const float* W2       = (const float*)d_in[5];
    const float* b2       = (const float*)d_in[6];

    const int n = in_sizes[0] / DFEAT;   // 100000
    const int e = in_sizes[1];           // 800000

    float* wsf      = (float*)d_ws;
    float* norm_src = wsf;                        // [n]  (holds deg_out first)
    float* norm_dst = wsf + n;                    // [n]  (holds deg_in first)
    float* h        = wsf + 2 * (size_t)n;        // [n,256]
    float* agg      = h + (size_t)n * DFEAT;      // [n,256]
    float* out      = (float*)d_out;              // [n,256]

    const int nb  = (n + 255) / 256;
    const int eb  = (e + 255) / 256;
    const dim3 gemm_grid((n + 15) / 16, DFEAT / 64);
    const int sc_blocks = (e + 3) / 4;
    const long total4   = (long)n * (DFEAT / 4);
    const int tb        = (int)((total4 + 255) / 256);

    // degrees -> norms
    gcn_init_deg <<<nb, 256, 0, stream>>>(norm_src, norm_dst, n);
    gcn_count_deg<<<eb, 256, 0, stream>>>(src, dst, norm_src, norm_dst, e);
    gcn_make_norm<<<nb, 256, 0, stream>>>(norm_src, norm_dst, n);

    // layer 1
    gcn_gemm_wmma<<<gemm_grid, 128, 0, stream>>>(features, norm_src, W1, h, agg, n);
    gcn_scatter  <<<sc_blocks, 256, 0, stream>>>(h, agg, src, dst, e);
    gcn_bias_act <<<tb, 256, 0, stream>>>(agg, norm_dst, b1, 1, total4);

    // layer 2 (scatter accumulates straight into d_out, seeded by gemm epilogue)
    gcn_gemm_wmma<<<gemm_grid, 128, 0, stream>>>(agg, norm_src, W2, h, out, n);
    gcn_scatter  <<<sc_blocks, 256, 0, stream>>>(h, out, src, dst, e);
    gcn_bias_act <<<tb, 256, 0, stream>>>(out, norm_dst, b2, 0, total4);
}